// GATv2MiniModel_42004780155064
// MI455X (gfx1250) — compile-verified
//
#include <hip/hip_runtime.h>
#include <hip/hip_bf16.h>
#include <math.h>

// ---------------- problem constants (match reference) ----------------
#define NU 50000
#define NI 50000
#define NE 400000
// H=4 heads, C=64 channels, H*C=256, HID=64

typedef __attribute__((ext_vector_type(2))) float v2f;
typedef __attribute__((ext_vector_type(8))) float v8f;

// Monotone float <-> ordered-uint mapping for atomicMax on floats.
__device__ __forceinline__ unsigned f2ord(float f) {
    unsigned u = __float_as_uint(f);
    return (u & 0x80000000u) ? ~u : (u | 0x80000000u);
}
__device__ __forceinline__ float ord2f(unsigned u) {
    return (u & 0x80000000u) ? __uint_as_float(u & 0x7fffffffu) : __uint_as_float(~u);
}

// ---------------------------------------------------------------------
// out = act(A[M,K] @ W[K,N] + bias), exact fp32 via V_WMMA_F32_16X16X4_F32.
// grid = (M/16, N/64), block = 128 threads (4 waves; wave w -> cols w*16..w*16+15
// of this block's 64-col slice). M,N multiples of 16/64, K multiple of 4.
// ---------------------------------------------------------------------
__global__ void gemm_wmma(const float* __restrict__ A, const float* __restrict__ W,
                          const float* __restrict__ bias, float* __restrict__ out,
                          int K, int N, int act)
{
    __shared__ float As[16 * 129];   // 16 x K tile, padded stride K+1
    __shared__ float Ws[128 * 65];   // K x 64 slice, padded stride 65
    const int m0  = blockIdx.x * 16;
    const int n0  = blockIdx.y * 64;
    const int tid = threadIdx.x;
    const int KP  = K + 1;

    for (int i = tid; i < 16 * K; i += 128) {
        int r = i / K, c = i - r * K;
        As[r * KP + c] = A[(size_t)(m0 + r) * K + c];
    }
    for (int i = tid; i < K * 64; i += 128) {
        int r = i >> 6, c = i & 63;
        Ws[r * 65 + c] = W[(size_t)r * N + n0 + c];
    }
    __syncthreads();

    const int lane = tid & 31;
    const int g    = lane >> 4;       // half-wave group
    const int r    = lane & 15;       // row (A) / col (B,C) within tile
    const int col  = ((tid >> 5) << 4) + r;   // column within 64-col slice

    v8f acc = {};
    for (int k = 0; k < K; k += 4) {
        v2f a, b;
        a.x = As[r * KP + k + 2 * g];
        a.y = As[r * KP + k + 2 * g + 1];
        b.x = Ws[(k + 2 * g) * 65 + col];
        b.y = Ws[(k + 2 * g + 1) * 65 + col];
        acc = __builtin_amdgcn_wmma_f32_16x16x4_f32(false, a, false, b,
                                                    (short)0, acc, false, false);
    }

    const int gcol = n0 + col;
    const float bv = bias ? bias[gcol] : 0.0f;
#pragma unroll
    for (int v = 0; v < 8; ++v) {
        float x = acc[v] + bv;
        if (act) x = fmaxf(x, 0.0f);
        out[(size_t)(m0 + v + 8 * g) * N + gcol] = x;
    }
}

// ---------------------------------------------------------------------
// Pass A: per edge, logit[e][h] = att_h . leakyrelu(xl[src]+xr[dst]+ea*We),
// and ordered-uint atomicMax into mbits[dst][h] (exact segment max).
// One wave per edge; lane covers channels lane, lane+32, ..., lane+224.
// ---------------------------------------------------------------------
__global__ void edge_logit(const float* __restrict__ xl, const float* __restrict__ xr,
                           const int* __restrict__ src, const int* __restrict__ dst,
                           const float* __restrict__ ea, const float* __restrict__ We,
                           const float* __restrict__ att,
                           float* __restrict__ logit, unsigned* __restrict__ mbits, int E)
{
    const int e    = blockIdx.x * 8 + (threadIdx.x >> 5);
    const int lane = threadIdx.x & 31;
    if (e >= E) return;
    const int s = src[e], d = dst[e];
    const float w = ea[e];
    const float* pl = xl + (size_t)s * 256;
    const float* pr = xr + (size_t)d * 256;

    float sum[4] = {0.f, 0.f, 0.f, 0.f};
#pragma unroll
    for (int j = 0; j < 8; ++j) {
        int c = lane + 32 * j;                 // head = j>>1 (compile-time)
        float v = pl[c] + pr[c] + w * We[c];
        v = (v > 0.f) ? v : 0.2f * v;          // leaky_relu(0.2)
        sum[j >> 1] += v * att[c];
    }
#pragma unroll
    for (int h = 0; h < 4; ++h) {
        float x = sum[h];
        for (int o = 16; o > 0; o >>= 1) x += __shfl_xor(x, o, 32);
        sum[h] = x;
    }
    if (lane == 0) {
#pragma unroll
        for (int h = 0; h < 4; ++h) {
            logit[(size_t)e * 4 + h] = sum[h];
            atomicMax(&mbits[(size_t)d * 4 + h], f2ord(sum[h]));
        }
    }
}

// Pass B: a = exp(logit - m[dst]); denom[dst][h] += a. One thread per (e,h).
__global__ void edge_exp(float* __restrict__ aexp, const int* __restrict__ dst,
                         const unsigned* __restrict__ mbits,
                         float* __restrict__ denom, int E4)
{
    const int i = blockIdx.x * 256 + threadIdx.x;   // i = e*4 + h
    if (i >= E4) return;
    const int e = i >> 2, h = i & 3;
    const int d = dst[e];
    const float m = ord2f(mbits[(size_t)d * 4 + h]);
    const float a = expf(aexp[i] - m);
    aexp[i] = a;
    atomicAdd(&denom[(size_t)d * 4 + h], a);
}

// Pass C: outacc[dst][c] += 0.25 * sum_h alpha[e][h]*xl[src][h*64+c]
// (head-mean folded in -> only 64 atomics per edge). One wave per edge.
__global__ void edge_aggregate(const float* __restrict__ xl,
                               const int* __restrict__ src, const int* __restrict__ dst,
                               const float* __restrict__ aexp, const float* __restrict__ denom,
                               float* __restrict__ outacc, int E)
{
    const int e    = blockIdx.x * 8 + (threadIdx.x >> 5);
    const int lane = threadIdx.x & 31;
    if (e >= E) return;
    const int s = src[e], d = dst[e];

    float alpha[4];
#pragma unroll
    for (int h = 0; h < 4; ++h)
        alpha[h] = aexp[(size_t)e * 4 + h] / (denom[(size_t)d * 4 + h] + 1e-16f);

    const float* pl = xl + (size_t)s * 256;
#pragma unroll
    for (int j = 0; j < 2; ++j) {
        int c = lane + 32 * j;
        float v = 0.f;
#pragma unroll
        for (int h = 0; h < 4; ++h) v += alpha[h] * pl[h * 64 + c];
        atomicAdd(&outacc[(size_t)d * 64 + c], 0.25f * v);
    }
}

// Pass D: h = relu(acc + bias[c])  (in place)
__global__ void bias_relu(float* __restrict__ h, const float* __restrict__ bias, int n)
{
    const int i = blockIdx.x * 256 + threadIdx.x;
    if (i < n) h[i] = fmaxf(h[i] + bias[i & 63], 0.0f);
}

// Row-wise L2 normalize (rows of 64), one wave per row, 8 rows per block.
__global__ void l2norm_rows(float* __restrict__ z, int nrows)
{
    const int row  = blockIdx.x * 8 + (threadIdx.x >> 5);
    const int lane = threadIdx.x & 31;
    if (row >= nrows) return;
    float* p = z + (size_t)row * 64;
    float a = p[lane], b = p[lane + 32];
    float s = a * a + b * b;
    for (int o = 16; o > 0; o >>= 1) s += __shfl_xor(s, o, 32);
    const float inv = 1.0f / fmaxf(sqrtf(s), 1e-12f);
    p[lane] = a * inv;
    p[lane + 32] = b * inv;
}

// ---------------------------------------------------------------------
static void run_conv(const float* hsrc, const float* hdst,
                     const int* ei, const float* ea,
                     const float* Wl, const float* bl, const float* Wr, const float* br,
                     const float* We, const float* att, const float* bias,
                     int Ns, int Nd, float* hout,
                     float* xl, float* xr, float* elog, float* denom, unsigned* mbits,
                     hipStream_t stream)
{
    const int* src = ei;        // row 0
    const int* dst = ei + NE;   // row 1
    gemm_wmma<<<dim3(Ns / 16, 4), 128, 0, stream>>>(hsrc, Wl, bl, xl, 64, 256, 0);
    gemm_wmma<<<dim3(Nd / 16, 4), 128, 0, stream>>>(hdst, Wr, br, xr, 64, 256, 0);
    hipMemsetAsync(mbits, 0, (size_t)Nd * 4 * sizeof(unsigned), stream);
    hipMemsetAsync(denom, 0, (size_t)Nd * 4 * sizeof(float), stream);
    hipMemsetAsync(hout,  0, (size_t)Nd * 64 * sizeof(float), stream);
    edge_logit<<<NE / 8, 256, 0, stream>>>(xl, xr, src, dst, ea, We, att, elog, mbits, NE);
    edge_exp<<<(NE * 4) / 256, 256, 0, stream>>>(elog, dst, mbits, denom, NE * 4);
    edge_aggregate<<<NE / 8, 256, 0, stream>>>(xl, src, dst, elog, denom, hout, NE);
    bias_relu<<<((size_t)Nd * 64 + 255) / 256, 256, 0, stream>>>(hout, bias, Nd * 64);
}

extern "C" void kernel_launch(void* const* d_in, const int* in_sizes, int n_in,
                              void* d_out, int out_size, void* d_ws, size_t ws_size,
                              hipStream_t stream)
{
    (void)in_sizes; (void)n_in; (void)out_size; (void)ws_size;

    const float* x_user = (const float*)d_in[0];
    const float* x_item = (const float*)d_in[1];
    const int*   ei_u2i = (const int*)d_in[2];
    const int*   ei_i2u = (const int*)d_in[3];
    const float* ea_u2i = (const float*)d_in[4];
    const float* ea_i2u = (const float*)d_in[5];
    const float* Wiu = (const float*)d_in[6];
    const float* biu = (const float*)d_in[7];
    const float* Wii = (const float*)d_in[8];
    const float* bii = (const float*)d_in[9];

    // ---- workspace layout (floats) ----
    float* ws = (float*)d_ws;
    size_t off = 0;
    float* hu[2]; float* hi[2];
    hu[0] = ws + off; off += (size_t)NU * 64;
    hu[1] = ws + off; off += (size_t)NU * 64;
    hi[0] = ws + off; off += (size_t)NI * 64;
    hi[1] = ws + off; off += (size_t)NI * 64;
    float* xl    = ws + off; off += (size_t)50000 * 256;
    float* xr    = ws + off; off += (size_t)50000 * 256;
    float* elog  = ws + off; off += (size_t)NE * 4;
    float* denom = ws + off; off += (size_t)50000 * 4;
    unsigned* mbits = (unsigned*)(ws + off); off += (size_t)50000 * 4;

    // ---- input projections: hu = relu(x_user@W+b), hi = relu(x_item@W+b) ----
    gemm_wmma<<<dim3(NU / 16, 1), 128, 0, stream>>>(x_user, Wiu, biu, hu[0], 128, 64, 1);
    gemm_wmma<<<dim3(NI / 16, 1), 128, 0, stream>>>(x_item, Wii, bii, hi[0], 64, 64, 1);

    // ---- 2 GATv2 layers, both directions read the same (hu,hi) then swap ----
    int cur = 0;
    for (int l = 0; l < 2; ++l) {
        const int bu = 10 + l * 14;   // l{l}_u2i params: Wl,bl,Wr,br,We,att,bias
        const int bi = bu + 7;        // l{l}_i2u params
        // u2i: src=user, dst=item -> writes hi[1-cur]
        run_conv(hu[cur], hi[cur], ei_u2i, ea_u2i,
                 (const float*)d_in[bu + 0], (const float*)d_in[bu + 1],
                 (const float*)d_in[bu + 2], (const float*)d_in[bu + 3],
                 (const float*)d_in[bu + 4], (const float*)d_in[bu + 5],
                 (const float*)d_in[bu + 6],
                 NU, NI, hi[1 - cur], xl, xr, elog, denom, mbits, stream);
        // i2u: src=item, dst=user -> writes hu[1-cur]
        run_conv(hi[cur], hu[cur], ei_i2u, ea_i2u,
                 (const float*)d_in[bi + 0], (const float*)d_in[bi + 1],
                 (const float*)d_in[bi + 2], (const float*)d_in[bi + 3],
                 (const float*)d_in[bi + 4], (const float*)d_in[bi + 5],
                 (const float*)d_in[bi + 6],
                 NI, NU, hu[1 - cur], xl, xr, elog, denom, mbits, stream);
        cur ^= 1;
    }

    // ---- final projections + row L2 norm, straight into d_out ----
    float* zu = (float*)d_out;
    float* zi = zu + (size_t)NU * 64;
    gemm_wmma<<<dim3(NU / 16, 1), 128, 0, stream>>>(hu[cur], (const float*)d_in[38],
                                                    (const float*)d_in[39], zu, 64, 64, 0);
    gemm_wmma<<<dim3(NI / 16, 1), 128, 0, stream>>>(hi[cur], (const float*)d_in[40],
                                                    (const float*)d_in[41], zi, 64, 64, 0);
    l2norm_rows<<<NU / 8, 256, 0, stream>>>(zu, NU);
    l2norm_rows<<<NI / 8, 256, 0, stream>>>(zi, NI);
}